// _LiftSplat_37048387895637
// MI455X (gfx1250) — compile-verified
//
#include <hip/hip_runtime.h>
#include <math.h>

typedef __attribute__((ext_vector_type(2))) float v2f;
typedef __attribute__((ext_vector_type(8))) float v8f;

// Fixed problem shape from the reference setup.
#define BB   2
#define TT   2
#define CC   6
#define CR   64
#define HP   32
#define WP   88
#define NP   (HP * WP)        // 2816 pixels per camera view
#define NSEG (BB * TT * CC)   // 24
#define NBT  (BB * TT)        // 4
#define HB_  200
#define WB_  200
#define NBIN (HB_ * WB_)      // 40000
#define XMIN (-50.0f)
#define YMIN (-50.0f)
#define INV_RX 2.0f           // 1 / ((X_MAX-X_MIN)/WB) = 1/0.5
#define INV_RY 2.0f

// Workspace layout (floats): [0 .. 288) per-segment params, [512 .. 512+NSEG*NBIN) counts
#define PRM_OFF 0
#define CNT_OFF 512

// ---------------------------------------------------------------------------
// Kernel 0: zero output accumulator + per-camera bin counts (poisoned memory).
__global__ void _ls_zero(float* __restrict__ out, float* __restrict__ cnt,
                         int n_out, int n_cnt) {
  int i = blockIdx.x * blockDim.x + threadIdx.x;
  if (i < n_out) out[i] = 0.0f;
  else if (i < n_out + n_cnt) cnt[i - n_out] = 0.0f;
}

// ---------------------------------------------------------------------------
// Kernel 1: per-segment constants. T_cam_from_ego is rigid ([R|t]), so
// inv = [R^T | -R^T t] exactly. Keep only rows 0,1 (x,y) + scaled intrinsics.
__global__ void _ls_prep(const float* __restrict__ K9,
                         const float* __restrict__ E16,
                         const int* __restrict__ Hptr,
                         const int* __restrict__ Wptr,
                         float* __restrict__ prm) {
  int n = threadIdx.x;
  if (n >= NSEG) return;
  const float Sx = (float)WP / (float)(*Wptr);
  const float Sy = (float)HP / (float)(*Hptr);
  const float* e = E16 + n * 16;
  const float r00 = e[0], r01 = e[1], r02 = e[2],  t0 = e[3];
  const float r10 = e[4], r11 = e[5], r12 = e[6],  t1 = e[7];
  const float r20 = e[8], r21 = e[9], r22 = e[10], t2 = e[11];
  (void)r02; (void)r12; (void)r22;
  float* p = prm + n * 12;
  // inv row 0 = [R00,R10,R20, -(R00 t0 + R10 t1 + R20 t2)]
  p[0] = r00; p[1] = r10; p[2] = r20; p[3] = -(r00 * t0 + r10 * t1 + r20 * t2);
  // inv row 1
  p[4] = r01; p[5] = r11; p[6] = r21; p[7] = -(r01 * t0 + r11 * t1 + r21 * t2);
  const float* k = K9 + n * 9;
  p[8]  = k[0] * Sx;  // fx'
  p[9]  = k[4] * Sy;  // fy'
  p[10] = k[2] * Sx;  // cx'
  p[11] = k[5] * Sy;  // cy'
}

// ---------------------------------------------------------------------------
// Kernel 2: project via V_WMMA_F32_16X16X4_F32 (16 pixels / wave / WMMA),
// then scatter 64 channels per pixel with pixel-coalesced feature loads.
// Block = 256 threads = 8 waves; each block covers 128 pixels of one segment.
__global__ void _ls_scatter(const float* __restrict__ feats,
                            const float* __restrict__ depths,
                            const int* __restrict__ Hptr,
                            const int* __restrict__ Wptr,
                            const float* __restrict__ prm,
                            float* __restrict__ cnt,
                            float* __restrict__ out) {
  const int n    = blockIdx.y;        // segment (b,t,c)
  const int bt   = n / CC;            // (b,t)
  const int lane = threadIdx.x & 31;
  const int wv   = threadIdx.x >> 5;
  const int p    = lane & 15;         // pixel slot within the 16-pixel group
  const int half = lane >> 4;         // 0: K=0,1 operands  1: K=2,3 operands
  const int pixbase = blockIdx.x * 128 + wv * 16;
  const int pix  = pixbase + p;       // both halves work on the same pixel
  const int vrow = pix / WP;
  const int ucol = pix - vrow * WP;

  // nearest-resize depth lookup: iy = int(v * H/Hp), ix = int(u * W/Wp)
  const int Hs = *Hptr, Ws = *Wptr;
  const int dy = (int)((float)vrow * ((float)Hs / (float)HP));
  const int dx = (int)((float)ucol * ((float)Ws / (float)WP));
  const float d = depths[((size_t)n * Hs + dy) * Ws + dx];

  const float* pr = prm + n * 12;
  const float fx = pr[8], fy = pr[9], cx = pr[10], cy = pr[11];
  const float xc = ((float)ucol - cx) / fx * d;
  const float yc = ((float)vrow - cy) / fy * d;

  // A: rows 0/1 = inverse-extrinsic rows (x,y), rows 2..15 = 0.
  //   lanes 0-15 hold K=0,1; lanes 16-31 hold K=2,3 (ISA 16x4 f32 A layout).
  // B: column N = pixel N's [xc, yc, d, 1]; same half-lane K split.
  v2f a; a.x = 0.0f; a.y = 0.0f;
  v2f b;
  if (half == 0) {
    if (p == 0)      { a.x = pr[0]; a.y = pr[1]; }
    else if (p == 1) { a.x = pr[4]; a.y = pr[5]; }
    b.x = xc; b.y = yc;                 // K=0,1 of this pixel's column
  } else {
    if (p == 0)      { a.x = pr[2]; a.y = pr[3]; }
    else if (p == 1) { a.x = pr[6]; a.y = pr[7]; }
    b.x = d;  b.y = 1.0f;               // K=2,3
  }

  v8f acc = {};
  // D = A(16x4) x B(4x16): lanes 0-15 get x (VGPR0, M=0) and y (VGPR1, M=1)
  acc = __builtin_amdgcn_wmma_f32_16x16x4_f32(
      /*neg_a=*/false, a, /*neg_b=*/false, b,
      /*c_mod=*/(short)0, acc, /*reuse_a=*/false, /*reuse_b=*/false);

  // Broadcast this pixel's (x,y) to both half-waves.
  const float x = __shfl(acc[0], p, 32);
  const float y = __shfl(acc[1], p, 32);

  const bool valid = (d > 0.0f) && (x >= XMIN) && (x < -XMIN) &&
                                   (y >= YMIN) && (y < -YMIN);
  int ix = (int)floorf((x - XMIN) * INV_RX);
  int iy = (int)floorf((y - YMIN) * INV_RY);
  ix = min(max(ix, 0), WB_ - 1);
  iy = min(max(iy, 0), HB_ - 1);
  const int bin = iy * WB_ + ix;

  if (valid && half == 0) atomicAdd(&cnt[(size_t)n * NBIN + bin], 1.0f);

  // Scatter: iteration t -> lane handles channel ch = 2t + half of pixel p.
  // feats loads are coalesced across lanes (16 consecutive pixels per half).
  const float* fp = feats + (size_t)n * CR * NP + pixbase + p;
  float* ob = out + (size_t)bt * CR * NBIN + bin;
  if (valid) {
#pragma unroll
    for (int t = 0; t < 32; ++t) {
      const int ch = 2 * t + half;
      atomicAdd(ob + (size_t)ch * NBIN, fp[(size_t)ch * NP]);
    }
  }
}

// ---------------------------------------------------------------------------
// Kernel 3: out[bt,ch,bin] /= sum_c max(cnt[bt,c,bin], 1)
__global__ void _ls_finalize(const float* __restrict__ cnt,
                             float* __restrict__ out) {
  const int g = blockIdx.x * blockDim.x + threadIdx.x;
  if (g >= NBT * NBIN) return;
  const int bt  = g / NBIN;
  const int bin = g - bt * NBIN;
  float den = 0.0f;
#pragma unroll
  for (int c = 0; c < CC; ++c)
    den += fmaxf(cnt[(size_t)(bt * CC + c) * NBIN + bin], 1.0f);
  const float inv = 1.0f / den;
  float* ob = out + (size_t)bt * CR * NBIN + bin;
#pragma unroll
  for (int ch = 0; ch < CR; ++ch)
    ob[(size_t)ch * NBIN] *= inv;   // coalesced across lanes per channel
}

// ---------------------------------------------------------------------------
extern "C" void kernel_launch(void* const* d_in, const int* in_sizes, int n_in,
                              void* d_out, int out_size, void* d_ws, size_t ws_size,
                              hipStream_t stream) {
  const float* feats  = (const float*)d_in[0];
  const float* depths = (const float*)d_in[1];
  const float* K9     = (const float*)d_in[2];
  const float* E16    = (const float*)d_in[3];
  const int*   Hptr   = (const int*)d_in[4];
  const int*   Wptr   = (const int*)d_in[5];
  float* out = (float*)d_out;
  float* ws  = (float*)d_ws;
  float* prm = ws + PRM_OFF;
  float* cnt = ws + CNT_OFF;

  const int n_out = NBT * CR * NBIN;   // 10,240,000
  const int n_cnt = NSEG * NBIN;       //    960,000
  _ls_zero<<<(n_out + n_cnt + 255) / 256, 256, 0, stream>>>(out, cnt, n_out, n_cnt);
  _ls_prep<<<1, 32, 0, stream>>>(K9, E16, Hptr, Wptr, prm);

  dim3 grid(NP / 128, NSEG);           // 22 x 24 blocks, 256 threads each
  _ls_scatter<<<grid, 256, 0, stream>>>(feats, depths, Hptr, Wptr, prm, cnt, out);

  _ls_finalize<<<(NBT * NBIN + 255) / 256, 256, 0, stream>>>(cnt, out);
}